// PermutationMatrix_16698832846901
// MI455X (gfx1250) — compile-verified
//
#include <hip/hip_runtime.h>
#include <math.h>

#define N      2048
#define TEMP   100.0f
#define ITERS  1000
#define NB     128     // persistent blocks for sinkhorn
#define TPB    256     // 8 waves per block (wave32)

typedef float v2f __attribute__((ext_vector_type(2)));
typedef float v8f __attribute__((ext_vector_type(8)));

// ---------------------------------------------------------------------------
// Kernel 1: K[r][j] = exp(100*(M[r][j]-rowmax_r)); also K^T; init v=1, ctrl=0
// ---------------------------------------------------------------------------
__global__ void __launch_bounds__(TPB) prep_kernel(const float* __restrict__ M,
                                                   float* __restrict__ Kb,
                                                   float* __restrict__ KT,
                                                   float* __restrict__ vvec,
                                                   unsigned* __restrict__ ctrl) {
  __shared__ float red[TPB / 32];
  const int row = blockIdx.x;
  const int tid = threadIdx.x;
  const float* mrow = M + (size_t)row * N;

  float vals[8];
  float mx = -INFINITY;
#pragma unroll
  for (int q = 0; q < 8; ++q) {
    vals[q] = mrow[tid + q * TPB];
    mx = fmaxf(mx, vals[q]);
  }
#pragma unroll
  for (int off = 16; off > 0; off >>= 1) mx = fmaxf(mx, __shfl_down(mx, off, 32));
  if ((tid & 31) == 0) red[tid >> 5] = mx;
  __syncthreads();
  mx = red[0];
#pragma unroll
  for (int w = 1; w < TPB / 32; ++w) mx = fmaxf(mx, red[w]);

#pragma unroll
  for (int q = 0; q < 8; ++q) {
    const int j = tid + q * TPB;
    const float e = expf(TEMP * (vals[q] - mx));
    Kb[(size_t)row * N + j] = e;
    KT[(size_t)j * N + row] = e;   // one-time transposed copy (keeps both matvecs coalesced)
  }
  if (row == 0) {
#pragma unroll
    for (int q = 0; q < 8; ++q) vvec[tid + q * TPB] = 1.0f;
    if (tid == 0) ctrl[0] = 0u;    // reset barrier counter every launch
  }
}

// ---------------------------------------------------------------------------
// Persistent Sinkhorn:  u = 1/(K v) ; barrier ; v = 1/(K^T u) ; barrier
// ---------------------------------------------------------------------------
__device__ __forceinline__ void grid_barrier(unsigned* ctrl, unsigned target) {
  __builtin_amdgcn_fence(__ATOMIC_RELEASE, "agent");  // flush this thread's stores
  __syncthreads();
  if (threadIdx.x == 0) {
    __hip_atomic_fetch_add(ctrl, 1u, __ATOMIC_RELEASE, __HIP_MEMORY_SCOPE_AGENT);
    while (__hip_atomic_load(ctrl, __ATOMIC_RELAXED, __HIP_MEMORY_SCOPE_AGENT) < target)
      __builtin_amdgcn_s_sleep(1);
  }
  __syncthreads();
  __builtin_amdgcn_fence(__ATOMIC_ACQUIRE, "agent");  // invalidate caches before re-reading
}

__device__ __forceinline__ void stage_vec(float* __restrict__ dst,
                                          const float* __restrict__ src) {
  const float4* s4 = (const float4*)src;
  float4* d4 = (float4*)dst;
#pragma unroll
  for (int q = 0; q < N / (4 * TPB); ++q)
    d4[threadIdx.x + q * TPB] = s4[threadIdx.x + q * TPB];
}

__device__ __forceinline__ void matvec_recip(const float* __restrict__ Mat,
                                             const float* __restrict__ sv,
                                             float* __restrict__ out,
                                             int gwave, int lane) {
#pragma unroll
  for (int rr = 0; rr < N / (NB * (TPB / 32)); ++rr) {   // 2 rows per wave
    const int row = gwave + rr * (NB * (TPB / 32));
    const float4* mr = (const float4*)(Mat + (size_t)row * N);
    const float4* v4 = (const float4*)sv;
    float acc = 0.f;
#pragma unroll 4
    for (int t = 0; t < N / (4 * 32); ++t) {             // 16 x float4 per lane
      const float4 a = mr[lane + t * 32];
      const float4 b = v4[lane + t * 32];
      acc = fmaf(a.x, b.x, acc);
      acc = fmaf(a.y, b.y, acc);
      acc = fmaf(a.z, b.z, acc);
      acc = fmaf(a.w, b.w, acc);
    }
#pragma unroll
    for (int off = 16; off > 0; off >>= 1) acc += __shfl_down(acc, off, 32);
    if (lane == 0) out[row] = 1.0f / acc;
  }
}

__global__ void __launch_bounds__(TPB, 1) sinkhorn_kernel(const float* __restrict__ Kb,
                                                          const float* __restrict__ KT,
                                                          float* __restrict__ u,
                                                          float* __restrict__ v,
                                                          unsigned* __restrict__ ctrl) {
  __shared__ __align__(16) float sv[N];   // 8KB staged broadcast vector
  const int tid = threadIdx.x;
  const int lane = tid & 31;
  const int gwave = blockIdx.x * (TPB / 32) + (tid >> 5);  // 0..1023
  unsigned seq = 0;

#pragma unroll 1
  for (int it = 0; it < ITERS; ++it) {
    stage_vec(sv, v);
    __syncthreads();
    matvec_recip(Kb, sv, u, gwave, lane);     // u = 1/(K v)   (row normalize)
    grid_barrier(ctrl, NB * (++seq));

    stage_vec(sv, u);
    __syncthreads();
    matvec_recip(KT, sv, v, gwave, lane);     // v = 1/(K^T u) (col normalize)
    grid_barrier(ctrl, NB * (++seq));
  }
}

// ---------------------------------------------------------------------------
// Kernel 3: m = diag(u) K diag(v) (in place over K);  P = row inclusive cumsum
// ---------------------------------------------------------------------------
__global__ void __launch_bounds__(TPB) scale_scan_kernel(float* __restrict__ Km,
                                                         const float* __restrict__ u,
                                                         const float* __restrict__ v,
                                                         float* __restrict__ P) {
  __shared__ float ss[TPB];
  const int row = blockIdx.x;
  const int tid = threadIdx.x;
  const float ui = u[row];
  const size_t base = (size_t)row * N + (size_t)tid * 8;

  float mm[8], inc[8];
  float run = 0.f;
#pragma unroll
  for (int q = 0; q < 8; ++q) {
    const float val = ui * Km[base + q] * v[tid * 8 + q];
    mm[q] = val;
    run += val;
    inc[q] = run;
  }
  ss[tid] = run;
  __syncthreads();
#pragma unroll
  for (int off = 1; off < TPB; off <<= 1) {   // Hillis-Steele inclusive scan
    float t = 0.f;
    if (tid >= off) t = ss[tid - off];
    __syncthreads();
    if (tid >= off) ss[tid] += t;
    __syncthreads();
  }
  const float pre = (tid == 0) ? 0.f : ss[tid - 1];
#pragma unroll
  for (int q = 0; q < 8; ++q) {
    Km[base + q] = mm[q];        // m (reused as GEMM B operand)
    P[base + q] = pre + inc[q];  // row prefix sum (GEMM A operand)
  }
}

// ---------------------------------------------------------------------------
// Kernel 4: O = P * m^T  via v_wmma_f32_16x16x4_f32 (f32 accumulate)
//   Block: 8 waves -> 32(i) x 64(j) tile, K-chunks of 32 staged in LDS.
// ---------------------------------------------------------------------------
__global__ void __launch_bounds__(TPB) gemm_kernel(const float* __restrict__ P,
                                                   const float* __restrict__ Mh,
                                                   float* __restrict__ Out) {
  __shared__ float Al[32][36];   // A chunk, padded rows (bank-conflict free b64 reads)
  __shared__ float Bl[32][65];   // B chunk transposed: Bl[k][j] = m[jb+j][kc+k]
  const int tid = threadIdx.x;
  const int lane = tid & 31;
  const int w = tid >> 5;        // wave 0..7
  const int iw = w & 1;          // i sub-tile 0..1
  const int jw = w >> 1;         // j sub-tile 0..3
  const int l16 = lane & 15;
  const int g = lane >> 4;       // lane group: supplies K = 2g, 2g+1
  const int ib = blockIdx.y * 32;
  const int jb = blockIdx.x * 64;

  v8f acc8 = {0.f, 0.f, 0.f, 0.f, 0.f, 0.f, 0.f, 0.f};

  for (int kc = 0; kc < N; kc += 32) {
    __syncthreads();
    {   // stage A: P[ib..ib+31][kc..kc+31]
      const int r = tid >> 3;
      const int c4 = (tid & 7) * 4;
      const float4 av = *(const float4*)(P + (size_t)(ib + r) * N + kc + c4);
      Al[r][c4 + 0] = av.x; Al[r][c4 + 1] = av.y;
      Al[r][c4 + 2] = av.z; Al[r][c4 + 3] = av.w;
    }
#pragma unroll
    for (int p = 0; p < 2; ++p) {  // stage B transposed: m[jb..jb+63][kc..kc+31]
      const int j = p * 32 + (tid >> 3);
      const int c4 = (tid & 7) * 4;
      const float4 bv = *(const float4*)(Mh + (size_t)(jb + j) * N + kc + c4);
      Bl[c4 + 0][j] = bv.x; Bl[c4 + 1][j] = bv.y;
      Bl[c4 + 2][j] = bv.z; Bl[c4 + 3][j] = bv.w;
    }
    __syncthreads();

#pragma unroll
    for (int kk = 0; kk < 32; kk += 4) {
      // A 16x4 f32: lane l -> M=l%16, VGPR r -> K = r + 2*(l/16)
      const v2f a = *(const v2f*)&Al[iw * 16 + l16][kk + 2 * g];
      // B 4x16 f32: VGPR r, lane group g -> K = r + 2g, N = l%16
      v2f b;
      b.x = Bl[kk + 2 * g + 0][jw * 16 + l16];
      b.y = Bl[kk + 2 * g + 1][jw * 16 + l16];
      acc8 = __builtin_amdgcn_wmma_f32_16x16x4_f32(false, a, false, b,
                                                   (short)0, acc8, false, false);
    }
  }

  // C/D 16x16 f32: VGPR r, lane l -> M = r + 8*(l/16), N = l%16
#pragma unroll
  for (int r = 0; r < 8; ++r)
    Out[(size_t)(ib + iw * 16 + r + 8 * g) * N + jb + jw * 16 + l16] = acc8[r];
}

// ---------------------------------------------------------------------------
extern "C" void kernel_launch(void* const* d_in, const int* in_sizes, int n_in,
                              void* d_out, int out_size, void* d_ws, size_t ws_size,
                              hipStream_t stream) {
  (void)in_sizes; (void)n_in; (void)out_size; (void)ws_size;
  const float* Min = (const float*)d_in[0];

  float* Kb = (float*)d_ws;                    // N*N : K, later m (in place)
  float* KT = Kb + (size_t)N * N;              // N*N : K^T, later P (reused)
  float* u  = KT + (size_t)N * N;              // N
  float* v  = u + N;                           // N
  unsigned* ctrl = (unsigned*)(v + N);         // barrier counter
  float* P  = KT;                              // KT dead after sinkhorn -> reuse as P
  float* Out = (float*)d_out;

  prep_kernel<<<N, TPB, 0, stream>>>(Min, Kb, KT, v, ctrl);
  sinkhorn_kernel<<<NB, TPB, 0, stream>>>(Kb, KT, u, v, ctrl);
  scale_scan_kernel<<<N, TPB, 0, stream>>>(Kb, u, v, P);
  gemm_kernel<<<dim3(N / 64, N / 32), TPB, 0, stream>>>(P, Kb, Out);
}